// EnhancedFeaturePropagation_57638461112795
// MI455X (gfx1250) — compile-verified
//
#include <hip/hip_runtime.h>

// Problem constants (fixed by the reference).
#define BB     8
#define PN     16384
#define NS     256
#define KDIM   512
#define HID    384
#define HALF   192
#define CIN1   (HID + HALF)   // 576
#define ROWS   (BB * NS)      // 2048
#define EPSBN  1e-5f

typedef float v2f __attribute__((ext_vector_type(2)));
typedef float v8f __attribute__((ext_vector_type(8)));

// ---------------------------------------------------------------------------
// 1) global_feat[b][o] = patch_feature[b,:] @ W_feat[:,o] + b_feat[o]
// ---------------------------------------------------------------------------
__global__ void gfeat_kernel(const float* __restrict__ pf,
                             const float* __restrict__ Wf,
                             const float* __restrict__ bf,
                             float* __restrict__ gfeat) {
  int i = blockIdx.x * blockDim.x + threadIdx.x;
  if (i >= BB * HID) return;
  int b = i / HID, o = i % HID;
  float acc = bf[o];
  for (int k = 0; k < KDIM; ++k)
    acc = fmaf(pf[b * KDIM + k], Wf[k * HID + o], acc);
  gfeat[i] = acc;
}

// ---------------------------------------------------------------------------
// 2) FPS in 3-D with 3x3 metric M = W_coord W_coord^T (bias cancels).
//    One block per batch; coords + running min-dist live in registers.
// ---------------------------------------------------------------------------
#define FPS_THREADS 1024
#define PPT (PN / FPS_THREADS)   // 16 points per thread

__global__ void __launch_bounds__(FPS_THREADS)
fps_kernel(const float* __restrict__ p, const float* __restrict__ Wc,
           int* __restrict__ selidx) {
  __shared__ float Msh[6];
  __shared__ float rv[32];
  __shared__ int   ri[32];
  __shared__ int   winner;

  const int b   = blockIdx.x;
  const int tid = threadIdx.x;
  const int lane = tid & 31;
  const int wid  = tid >> 5;

  if (tid == 0) {
    float m00 = 0.f, m01 = 0.f, m02 = 0.f, m11 = 0.f, m12 = 0.f, m22 = 0.f;
    for (int h = 0; h < HALF; ++h) {
      float w0 = Wc[0 * HALF + h], w1 = Wc[1 * HALF + h], w2 = Wc[2 * HALF + h];
      m00 = fmaf(w0, w0, m00); m01 = fmaf(w0, w1, m01); m02 = fmaf(w0, w2, m02);
      m11 = fmaf(w1, w1, m11); m12 = fmaf(w1, w2, m12); m22 = fmaf(w2, w2, m22);
    }
    Msh[0] = m00; Msh[1] = m01; Msh[2] = m02; Msh[3] = m11; Msh[4] = m12; Msh[5] = m22;
  }
  __syncthreads();
  const float m00 = Msh[0], m01 = Msh[1], m02 = Msh[2];
  const float m11 = Msh[3], m12 = Msh[4], m22 = Msh[5];

  float px[PPT], py[PPT], pz[PPT], dd[PPT];
  #pragma unroll
  for (int j = 0; j < PPT; ++j) {
    int i = tid + j * FPS_THREADS;
    const float* pp = p + ((size_t)b * PN + i) * 3;
    px[j] = pp[0]; py[j] = pp[1]; pz[j] = pp[2];
    dd[j] = 1e10f;
  }

  int far = 0;
  for (int t = 0; t < NS; ++t) {
    if (tid == 0) selidx[b * NS + t] = far;
    const float* cp = p + ((size_t)b * PN + far) * 3;
    float cx = cp[0], cy = cp[1], cz = cp[2];

    float bv = -1.f; int bi = 0x7fffffff;
    #pragma unroll
    for (int j = 0; j < PPT; ++j) {
      float dx = px[j] - cx, dy = py[j] - cy, dz = pz[j] - cz;
      float d = m00 * dx * dx + m11 * dy * dy + m22 * dz * dz
              + 2.f * (m01 * dx * dy + m02 * dx * dz + m12 * dy * dz);
      float v = fminf(dd[j], d);
      dd[j] = v;
      int i = tid + j * FPS_THREADS;
      if (v > bv || (v == bv && i < bi)) { bv = v; bi = i; }
    }

    // intra-wave argmax (wave32)
    #pragma unroll
    for (int off = 16; off > 0; off >>= 1) {
      float ov = __shfl_down(bv, off, 32);
      int   oi = __shfl_down(bi, off, 32);
      if (ov > bv || (ov == bv && oi < bi)) { bv = ov; bi = oi; }
    }
    if (lane == 0) { rv[wid] = bv; ri[wid] = bi; }
    __syncthreads();
    if (tid < 32) {
      float v2 = rv[tid]; int i2 = ri[tid];
      #pragma unroll
      for (int off = 16; off > 0; off >>= 1) {
        float ov = __shfl_down(v2, off, 32);
        int   oi = __shfl_down(i2, off, 32);
        if (ov > v2 || (ov == v2 && oi < i2)) { v2 = ov; i2 = oi; }
      }
      if (tid == 0) winner = i2;
    }
    __syncthreads();
    far = winner;
  }
}

// ---------------------------------------------------------------------------
// 3) combined[b,n,:] = [gfeat[b,:], p[b,sel[b,n],:] @ W_coord + b_coord]
// ---------------------------------------------------------------------------
__global__ void build_x_kernel(const float* __restrict__ gfeat,
                               const float* __restrict__ p,
                               const int* __restrict__ selidx,
                               const float* __restrict__ Wc,
                               const float* __restrict__ bc,
                               float* __restrict__ X) {
  int idx = blockIdx.x * blockDim.x + threadIdx.x;
  if (idx >= ROWS * CIN1) return;
  int c = idx % CIN1, r = idx / CIN1;
  int b = r / NS, n = r % NS;
  float v;
  if (c < HID) {
    v = gfeat[b * HID + c];
  } else {
    int cc = c - HID;
    int pi = selidx[b * NS + n];
    const float* pp = p + ((size_t)b * PN + pi) * 3;
    v = fmaf(pp[0], Wc[0 * HALF + cc],
        fmaf(pp[1], Wc[1 * HALF + cc],
        fmaf(pp[2], Wc[2 * HALF + cc], bc[cc])));
  }
  X[idx] = v;
}

// ---------------------------------------------------------------------------
// 4) fp32 WMMA GEMM: D[m,n] = sum_k A[m,k] * Wt[n,k] + bias[n]
//    A: Mrows x K row-major; Wt: Ncols x K row-major (PyTorch (out,in) layout).
//    One wave computes a 16x64 output strip (4 N-tiles):
//      - A fragment loaded once per k-step, reused by 4 WMMAs (4x less A traffic)
//      - 4 independent accumulator chains (spaces out WMMA RAW hazards)
//      - register double-buffer: k+4 fragments prefetched while k computes,
//        so the backend can issue partial s_wait_loadcnt instead of draining
//        the load counter before every WMMA.
// ---------------------------------------------------------------------------
#define NSTRIP 4

__global__ void wmma_gemm_f32(const float* __restrict__ A,
                              const float* __restrict__ Wt,
                              const float* __restrict__ bias,
                              float* __restrict__ D,
                              int Ncols, int K) {
  const int wave    = (blockIdx.x * blockDim.x + threadIdx.x) >> 5;
  const int nstrips = Ncols >> 6;          // strips of 64 columns
  const int tm = wave / nstrips;
  const int ts = wave % nstrips;
  const int lane = threadIdx.x & 31;
  const int ln = lane & 15;
  const int hf = lane >> 4;

  const float* arow = A  + (size_t)(tm * 16 + ln) * K + 2 * hf;
  const float* brow = Wt + (size_t)(ts * 64 + ln) * K + 2 * hf;
  const size_t bstride = (size_t)16 * K;   // next N-tile of the strip

  v8f acc[NSTRIP];
  #pragma unroll
  for (int j = 0; j < NSTRIP; ++j) acc[j] = (v8f){};

  // prologue: load k=0 fragments
  v2f a_c = *(const v2f*)(arow);
  v2f b_c[NSTRIP];
  #pragma unroll
  for (int j = 0; j < NSTRIP; ++j)
    b_c[j] = *(const v2f*)(brow + (size_t)j * bstride);

  // steady state: prefetch k+4 while computing k
  for (int k = 0; k + 4 < K; k += 4) {
    v2f a_n = *(const v2f*)(arow + k + 4);
    v2f b_n[NSTRIP];
    #pragma unroll
    for (int j = 0; j < NSTRIP; ++j)
      b_n[j] = *(const v2f*)(brow + (size_t)j * bstride + k + 4);

    #pragma unroll
    for (int j = 0; j < NSTRIP; ++j)
      acc[j] = __builtin_amdgcn_wmma_f32_16x16x4_f32(
          false, a_c, false, b_c[j], (short)0, acc[j], false, false);

    a_c = a_n;
    #pragma unroll
    for (int j = 0; j < NSTRIP; ++j) b_c[j] = b_n[j];
  }
  // epilogue: last k-step
  #pragma unroll
  for (int j = 0; j < NSTRIP; ++j)
    acc[j] = __builtin_amdgcn_wmma_f32_16x16x4_f32(
        false, a_c, false, b_c[j], (short)0, acc[j], false, false);

  // write back: C/D layout -> VGPR r holds M=r (lanes 0-15) / M=r+8 (lanes 16-31)
  #pragma unroll
  for (int j = 0; j < NSTRIP; ++j) {
    const int n = ts * 64 + j * 16 + ln;
    const float bv = bias[n];
    #pragma unroll
    for (int r = 0; r < 8; ++r) {
      int m = tm * 16 + r + 8 * hf;
      D[(size_t)m * Ncols + n] = acc[j][r] + bv;
    }
  }
}

// ---------------------------------------------------------------------------
// 5) BN stats (train mode, biased var over 2048 rows per channel)
//    -> scale = g * rsqrt(var+eps), shift = beta - mean*scale
// ---------------------------------------------------------------------------
__global__ void bn_stats_kernel(const float* __restrict__ H,
                                const float* __restrict__ g,
                                const float* __restrict__ be,
                                float* __restrict__ scale,
                                float* __restrict__ shift, int C) {
  __shared__ float ssum[256], ssq[256];
  const int ch = blockIdx.x;
  const int tid = threadIdx.x;
  float s = 0.f, q = 0.f;
  for (int r = tid; r < ROWS; r += 256) {
    float v = H[(size_t)r * C + ch];
    s += v; q = fmaf(v, v, q);
  }
  ssum[tid] = s; ssq[tid] = q;
  __syncthreads();
  for (int off = 128; off > 0; off >>= 1) {
    if (tid < off) { ssum[tid] += ssum[tid + off]; ssq[tid] += ssq[tid + off]; }
    __syncthreads();
  }
  if (tid == 0) {
    float mean = ssum[0] * (1.f / ROWS);
    float var  = ssq[0] * (1.f / ROWS) - mean * mean;
    float inv  = rsqrtf(var + EPSBN);
    float sc   = g[ch] * inv;
    scale[ch] = sc;
    shift[ch] = be[ch] - mean * sc;
  }
}

// ---------------------------------------------------------------------------
// 6) y = relu(x * scale[c] + shift[c])
// ---------------------------------------------------------------------------
__global__ void bn_apply_kernel(const float* __restrict__ in,
                                const float* __restrict__ scale,
                                const float* __restrict__ shift,
                                float* __restrict__ out, int total, int C) {
  int i = blockIdx.x * blockDim.x + threadIdx.x;
  if (i >= total) return;
  int c = i % C;
  float v = fmaf(in[i], scale[c], shift[c]);
  out[i] = v > 0.f ? v : 0.f;
}

// ---------------------------------------------------------------------------
extern "C" void kernel_launch(void* const* d_in, const int* in_sizes, int n_in,
                              void* d_out, int out_size, void* d_ws, size_t ws_size,
                              hipStream_t stream) {
  const float* p      = (const float*)d_in[0];
  // d_in[1] = N (256), fixed by reference constants
  const float* pf     = (const float*)d_in[2];
  const float* W_feat = (const float*)d_in[3];
  const float* b_feat = (const float*)d_in[4];
  const float* W_c    = (const float*)d_in[5];
  const float* b_c    = (const float*)d_in[6];
  const float* W1     = (const float*)d_in[7];
  const float* b1     = (const float*)d_in[8];
  const float* g1     = (const float*)d_in[9];
  const float* be1    = (const float*)d_in[10];
  const float* W2     = (const float*)d_in[11];
  const float* b2     = (const float*)d_in[12];
  const float* g2     = (const float*)d_in[13];
  const float* be2    = (const float*)d_in[14];
  float* out = (float*)d_out;

  // Workspace layout (floats); all offsets 16-float aligned.
  float* ws     = (float*)d_ws;
  float* gfeat  = ws;                       // 3072
  int*   selidx = (int*)(ws + 4096);        // 2048 ints
  float* X      = ws + 8192;                // 2048*576
  float* h1     = X  + (size_t)ROWS * CIN1; // 2048*384
  float* h2     = h1 + (size_t)ROWS * HID;  // 2048*384
  float* s1     = h2 + (size_t)ROWS * HID;
  float* t1     = s1 + HID;
  float* s2     = t1 + HID;
  float* t2     = s2 + HID;

  // 1) global features
  gfeat_kernel<<<(BB * HID + 255) / 256, 256, 0, stream>>>(pf, W_feat, b_feat, gfeat);

  // 2) farthest-point sampling via 3x3 metric (one block per batch)
  fps_kernel<<<BB, FPS_THREADS, 0, stream>>>(p, W_c, selidx);

  // 3) build combined matrix (2048 x 576)
  build_x_kernel<<<(ROWS * CIN1 + 255) / 256, 256, 0, stream>>>(
      gfeat, p, selidx, W_c, b_c, X);

  // 4) h1 = X @ W1^T + b1   (128 M-tiles x 6 N-strips = 768 waves -> 96 blocks)
  wmma_gemm_f32<<<(ROWS / 16) * (HID / 64) / 8, 256, 0, stream>>>(
      X, W1, b1, h1, HID, CIN1);

  // 5) BN1 stats + apply (relu), in place
  bn_stats_kernel<<<HID, 256, 0, stream>>>(h1, g1, be1, s1, t1, HID);
  bn_apply_kernel<<<(ROWS * HID + 255) / 256, 256, 0, stream>>>(
      h1, s1, t1, h1, ROWS * HID, HID);

  // 6) h2 = h1 @ W2^T + b2
  wmma_gemm_f32<<<(ROWS / 16) * (HID / 64) / 8, 256, 0, stream>>>(
      h1, W2, b2, h2, HID, HID);

  // 7) BN2 stats + apply (relu) -> d_out
  bn_stats_kernel<<<HID, 256, 0, stream>>>(h2, g2, be2, s2, t2, HID);
  bn_apply_kernel<<<(ROWS * HID + 255) / 256, 256, 0, stream>>>(
      h2, s2, t2, out, ROWS * HID, HID);
}